// Net_10213432230095
// MI455X (gfx1250) — compile-verified
//
#include <hip/hip_runtime.h>
#include <math.h>

typedef __attribute__((ext_vector_type(2))) float v2f;
typedef __attribute__((ext_vector_type(8))) float v8f;

// ---------------------------------------------------------------------------
// Generic fp32 WMMA GEMM: C[M,Nc] = A[M,K] @ W[K,Nc] + bias
// Row-major A (lda), W (ldw), C (ldc). M%16==0, Nc%16==0, K%4==0.
// One wave per 16x16 output tile, V_WMMA_F32_16X16X4_F32 over K in steps of 4.
// ---------------------------------------------------------------------------
__global__ void wmma_gemm_bias(const float* __restrict__ A,
                               const float* __restrict__ W,
                               const float* __restrict__ bias,
                               float* __restrict__ C,
                               int M, int K, int Nc,
                               int lda, int ldw, int ldc) {
    const int ntn    = Nc >> 4;
    const int ntiles = (M >> 4) * ntn;
    const int wid    = threadIdx.x >> 5;
    const int lane   = threadIdx.x & 31;
    const int tile   = blockIdx.x * 4 + wid;
    if (tile >= ntiles) return;           // whole-wave uniform exit: EXEC stays all-1s

    const int tm  = tile / ntn;
    const int tn  = tile % ntn;
    const int hl  = lane >> 4;            // lane half (0/1)
    const int lid = lane & 15;
    const int row = tm * 16 + lid;        // A row held by this lane
    const int col = tn * 16 + lid;        // B/D column held by this lane
    const int kb  = 2 * hl;               // K sub-offset for this lane half

    const float bv = bias ? bias[col] : 0.0f;
    v8f acc = {bv, bv, bv, bv, bv, bv, bv, bv};

    const float* Arow = A + (long)row * lda;
    for (int k0 = 0; k0 < K; k0 += 4) {
        v2f af, bf;
        af.x = Arow[k0 + kb];
        af.y = Arow[k0 + kb + 1];
        bf.x = W[(long)(k0 + kb)     * ldw + col];
        bf.y = W[(long)(k0 + kb + 1) * ldw + col];
        acc = __builtin_amdgcn_wmma_f32_16x16x4_f32(
                  false, af, false, bf, (short)0, acc, false, false);
    }

    // C/D layout: VGPR r -> row tm*16 + r + 8*half, col = lid
    #pragma unroll
    for (int r = 0; r < 8; ++r) {
        C[(long)(tm * 16 + r + 8 * hl) * ldc + col] = acc[r];
    }
}

// Copy x (rows x cols, dense) into a strided destination (concat buffer col 0)
__global__ void copy_strided(const float* __restrict__ src, float* __restrict__ dst,
                             int rows, int cols, int dst_stride) {
    int idx = blockIdx.x * blockDim.x + threadIdx.x;
    if (idx >= rows * cols) return;
    int r = idx / cols, c = idx - r * cols;
    dst[(long)r * dst_stride + c] = src[idx];
}

__device__ __forceinline__ float wave_sum32(float v) {
    #pragma unroll
    for (int off = 16; off > 0; off >>= 1) v += __shfl_xor(v, off, 32);
    return v;
}

// ---------------------------------------------------------------------------
// Row pass: one block per target node i. Recomputes stack[i,j,:] on the fly
// (c = lane 0..31 channel), computes att_i, accumulates agg_i[i,:] and writes
// e_out[i,j] = stack . ew + eb (layer 1 only).
// ---------------------------------------------------------------------------
__global__ void xenet_rowpass(const float* __restrict__ P, const float* __restrict__ Q,
                              const float* __restrict__ E, const float* __restrict__ Adj,
                              const float* __restrict__ w1, const float* __restrict__ w2,
                              const float* __restrict__ sb,
                              const float* __restrict__ aiw, const float* __restrict__ aib,
                              const float* __restrict__ ew, const float* __restrict__ eb,
                              float* __restrict__ agg_out, int agg_stride,
                              float* __restrict__ eout, int N) {
    const int i  = blockIdx.x;
    const int c  = threadIdx.x & 31;
    const int jw = threadIdx.x >> 5;      // wave index 0..7
    const float pc  = P[i * 32 + c];
    const float w1c = w1[c], w2c = w2[c], sbc = sb[c], aic = aiw[c];
    const float aib0 = aib[0];
    const float ewc = eout ? ew[c] : 0.0f;
    const float eb0 = eout ? eb[0] : 0.0f;

    float acc = 0.0f;
    for (int j = jw; j < N; j += 8) {
        float s = pc + Q[j * 32 + c] + E[(long)i * N + j] * w1c
                     + E[(long)j * N + i] * w2c + sbc;
        s = fmaxf(s, 0.0f) * Adj[(long)i * N + j];
        float d   = wave_sum32(s * aic);
        float att = 1.0f / (1.0f + __expf(-(d + aib0)));
        acc += att * s;
        if (eout) {
            float de = wave_sum32(s * ewc);
            if (c == 0) eout[(long)i * N + j] = de + eb0;
        }
    }
    __shared__ float red[8][32];
    red[jw][c] = acc;
    __syncthreads();
    if (jw == 0) {
        float t = 0.0f;
        #pragma unroll
        for (int w = 0; w < 8; ++w) t += red[w][c];
        agg_out[(long)i * agg_stride + c] = t;
    }
}

// ---------------------------------------------------------------------------
// Col pass: one block per source node j; att_j gate, accumulates agg_j[j,:].
// ---------------------------------------------------------------------------
__global__ void xenet_colpass(const float* __restrict__ P, const float* __restrict__ Q,
                              const float* __restrict__ E, const float* __restrict__ Adj,
                              const float* __restrict__ w1, const float* __restrict__ w2,
                              const float* __restrict__ sb,
                              const float* __restrict__ ajw, const float* __restrict__ ajb,
                              float* __restrict__ agg_out, int agg_stride, int N) {
    const int j  = blockIdx.x;
    const int c  = threadIdx.x & 31;
    const int iw = threadIdx.x >> 5;
    const float qc  = Q[j * 32 + c];
    const float w1c = w1[c], w2c = w2[c], sbc = sb[c], ajc = ajw[c];
    const float ajb0 = ajb[0];

    float acc = 0.0f;
    for (int i = iw; i < N; i += 8) {
        float s = P[i * 32 + c] + qc + E[(long)i * N + j] * w1c
                     + E[(long)j * N + i] * w2c + sbc;
        s = fmaxf(s, 0.0f) * Adj[(long)i * N + j];
        float d   = wave_sum32(s * ajc);
        float att = 1.0f / (1.0f + __expf(-(d + ajb0)));
        acc += att * s;
    }
    __shared__ float red[8][32];
    red[iw][c] = acc;
    __syncthreads();
    if (iw == 0) {
        float t = 0.0f;
        #pragma unroll
        for (int w = 0; w < 8; ++w) t += red[w][c];
        agg_out[(long)j * agg_stride + c] = t;
    }
}

// ---------------------------------------------------------------------------
extern "C" void kernel_launch(void* const* d_in, const int* in_sizes, int n_in,
                              void* d_out, int out_size, void* d_ws, size_t ws_size,
                              hipStream_t stream) {
    const int N = 400, F = 240, Ch = 32, NIN = F + 2 * Ch /*304*/, L = 1440;

    const float* x   = (const float*)d_in[0];
    const float* adj = (const float*)d_in[1];
    const float* e   = (const float*)d_in[2];   // (N,N,1) -> flat N*N

    const float* sw[2]  = {(const float*)d_in[3],  (const float*)d_in[13]};
    const float* sb[2]  = {(const float*)d_in[4],  (const float*)d_in[14]};
    const float* aiw[2] = {(const float*)d_in[5],  (const float*)d_in[15]};
    const float* aib[2] = {(const float*)d_in[6],  (const float*)d_in[16]};
    const float* ajw[2] = {(const float*)d_in[7],  (const float*)d_in[17]};
    const float* ajb[2] = {(const float*)d_in[8],  (const float*)d_in[18]};
    const float* nw[2]  = {(const float*)d_in[9],  (const float*)d_in[19]};
    const float* nb[2]  = {(const float*)d_in[10], (const float*)d_in[20]};
    const float* ew1    = (const float*)d_in[11];
    const float* eb1    = (const float*)d_in[12];
    const float* dw     = (const float*)d_in[23];
    const float* db     = (const float*)d_in[24];

    float* ws   = (float*)d_ws;
    float* P    = ws;                 // N*32
    float* Q    = P    + N * Ch;      // N*32
    float* xcat = Q    + N * Ch;      // N*304  concat [x | agg_i | agg_j]
    float* x2   = xcat + N * NIN;     // N*240  layer-1 node output
    float* e2   = x2   + N * F;       // N*N    layer-1 edge output
    float* x3   = e2   + N * N;       // N*240  layer-2 node output

    auto gemm = [&](const float* A, const float* W, const float* bias, float* Cout,
                    int M, int K, int Nc, int lda, int ldw, int ldc) {
        int ntiles = (M / 16) * (Nc / 16);
        int blocks = (ntiles + 3) / 4;
        wmma_gemm_bias<<<blocks, 128, 0, stream>>>(A, W, bias, Cout,
                                                   M, K, Nc, lda, ldw, ldc);
    };

    const float* xin = x;
    const float* ein = e;
    float* xouts[2] = {x2, x3};

    for (int layer = 0; layer < 2; ++layer) {
        const float* SW = sw[layer];
        const float* w1 = SW + (size_t)(2 * F)     * Ch;   // row for e
        const float* w2 = SW + (size_t)(2 * F + 1) * Ch;   // row for e^T
        // P = x @ sw[0:F], Q = x @ sw[F:2F]  (fp32 WMMA)
        gemm(xin, SW,              nullptr, P, N, F, Ch, F, Ch, Ch);
        gemm(xin, SW + (size_t)F * Ch, nullptr, Q, N, F, Ch, F, Ch, Ch);
        // concat buffer: [:,0:240] = x
        copy_strided<<<(N * F + 255) / 256, 256, 0, stream>>>(xin, xcat, N, F, NIN);
        // agg_i into xcat[:,240:272]; layer 0 also produces e_out -> e2
        xenet_rowpass<<<N, 256, 0, stream>>>(
            P, Q, ein, adj, w1, w2, sb[layer], aiw[layer], aib[layer],
            (layer == 0) ? ew1 : nullptr, (layer == 0) ? eb1 : nullptr,
            xcat + F, NIN, (layer == 0) ? e2 : nullptr, N);
        // agg_j into xcat[:,272:304]
        xenet_colpass<<<N, 256, 0, stream>>>(
            P, Q, ein, adj, w1, w2, sb[layer], ajw[layer], ajb[layer],
            xcat + F + Ch, NIN, N);
        // node model: x_out = xcat @ nw + nb
        gemm(xcat, nw[layer], nb[layer], xouts[layer], N, NIN, F, NIN, F, F);
        xin = xouts[layer];
        ein = e2;
    }

    // final dense: out = x3 @ dw + db  (400x240 @ 240x1440)
    gemm(x3, dw, db, (float*)d_out, N, F, L, F, L, L);
}